// TypeAwareAlignmentLoss_75007308858097
// MI455X (gfx1250) — compile-verified
//
#include <hip/hip_runtime.h>

typedef float v2f __attribute__((ext_vector_type(2)));
typedef float v8f __attribute__((ext_vector_type(8)));

#define NS 1500
#define NT 1500
#define NR 4
#define DD 64
#define ROWLEN (NR * DD)          // 256 floats per (row) = [k][d]
#define NB_B 200
#define EPSILON 3.0
#define OMEGA 0.4
#define RVAL 0.2

#define IPB 8                     // s-rows per block (one per wave32)
#define TCHUNK 48                 // t-rows staged in LDS per block iteration
#define KSTR 66                   // padded per-k stride (floats) in LDS
#define JSTR (KSTR * NR)          // 264 floats per LDS row
#define NBLOCKS ((NS + IPB - 1) / IPB)  // 188

// ---------------------------------------------------------------------------
// Kernel 1: anchor-pair L1 sums per type k -> ws[0..3]
// ---------------------------------------------------------------------------
__global__ __launch_bounds__(256) void anchor_kernel(
    const float* __restrict__ s, const float* __restrict__ t,
    const int* __restrict__ rows, const int* __restrict__ cols,
    float* __restrict__ ws) {
  __shared__ float red[256];
  const int tid = threadIdx.x;
  float accs[NR];
#pragma unroll
  for (int k = 0; k < NR; ++k) {
    float a = 0.0f;
    for (int e = tid; e < NB_B * DD; e += 256) {
      const int b = e >> 6;
      const int d = e & 63;
      const int r = rows[b];
      const int c = cols[b];
      a += __builtin_fabsf(s[r * ROWLEN + k * DD + d] -
                           t[c * ROWLEN + k * DD + d]);
    }
    accs[k] = a;
  }
  for (int k = 0; k < NR; ++k) {
    red[tid] = accs[k];
    __syncthreads();
    for (int sft = 128; sft > 0; sft >>= 1) {
      if (tid < sft) red[tid] += red[tid + sft];
      __syncthreads();
    }
    if (tid == 0) ws[k] = red[0];
    __syncthreads();
  }
}

// ---------------------------------------------------------------------------
// Kernel 2: all-pairs L1 sums per type k, block partials -> ws[4 + bid*4 + k]
// Wave w owns s-row i = bid*8 + w. t staged in LDS in TCHUNK tiles.
// WMMA f32 16x16x4 with B = ones reduces 4 d-values for 16 j-rows per issue,
// accumulating exactly in the f32 C matrix.
// ---------------------------------------------------------------------------
__global__ __launch_bounds__(256) void allpairs_kernel(
    const float* __restrict__ s, const float* __restrict__ t,
    float* __restrict__ ws) {
  __shared__ float t_lds[TCHUNK * JSTR];   // ~50 KB
  __shared__ float s_lds[IPB * JSTR];      // ~8 KB
  __shared__ float red[IPB][NR];

  const int tid  = threadIdx.x;
  const int wave = tid >> 5;
  const int lane = tid & 31;
  const int jj   = lane & 15;              // A-matrix row M this lane feeds
  const int off  = (lane >> 4) << 1;       // K offset: 0 (lanes 0-15) or 2

  const int i  = blockIdx.x * IPB + wave;
  const int ci = (i < NS) ? i : (NS - 1);

  // stage this wave's s row into padded LDS layout
  for (int q = lane; q < ROWLEN; q += 32) {
    const int k = q >> 6, d = q & 63;
    s_lds[wave * JSTR + k * KSTR + d] = s[ci * ROWLEN + q];
  }

  float acc[NR] = {0.f, 0.f, 0.f, 0.f};
  v2f bones;
  bones.x = 1.0f;
  bones.y = 1.0f;                          // B = ones(4x16), layout-agnostic

  for (int j0 = 0; j0 < NT; j0 += TCHUNK) {
    const int jcount = (NT - j0) < TCHUNK ? (NT - j0) : TCHUNK;

    // cooperative float2 load of the t tile into padded LDS
    const int nf2 = jcount * (ROWLEN / 2);
    for (int idx = tid; idx < nf2; idx += 256) {
      const int jl = idx / (ROWLEN / 2);
      const int q  = idx - jl * (ROWLEN / 2);
      const int k  = q >> 5;               // q / 32
      const int d2 = q & 31;
      const float2 v =
          ((const float2*)t)[(size_t)(j0 + jl) * (ROWLEN / 2) + q];
      *(float2*)&t_lds[jl * JSTR + k * KSTR + 2 * d2] = v;
    }
    __syncthreads();

    const int ngr = (jcount + 15) >> 4;    // 16-j groups (tail masked)
#pragma unroll
    for (int k = 0; k < NR; ++k) {
      v8f C = {};                          // exact f32 accumulator (8 VGPRs)
      for (int jg = 0; jg < ngr; ++jg) {
        const int jl = jg * 16 + jj;
        const bool jok = (jl < jcount);    // per-lane select, EXEC stays full
        const float* tp = &t_lds[jl * JSTR + k * KSTR + off];
        const float* sp = &s_lds[wave * JSTR + k * KSTR + off];
#pragma unroll
        for (int c = 0; c < 16; ++c) {     // d = 4*c + off + {0,1}
          const float2 tv = *(const float2*)(tp + 4 * c);
          const float2 sv = *(const float2*)(sp + 4 * c);
          v2f A;
          A.x = jok ? __builtin_fabsf(sv.x - tv.x) : 0.0f;
          A.y = jok ? __builtin_fabsf(sv.y - tv.y) : 0.0f;
          // D = A(16x4) * ones(4x16) + C  => row sums over 4 d-values
          C = __builtin_amdgcn_wmma_f32_16x16x4_f32(
              false, A, false, bones, (short)0, C, false, false);
        }
      }
      acc[k] += C[0] + C[1] + C[2] + C[3] + C[4] + C[5] + C[6] + C[7];
    }
    __syncthreads();                       // before next tile overwrites LDS
  }

  // lanes 0-15 hold sum of rows 0-7, lanes 16-31 rows 8-15 -> swap-add
#pragma unroll
  for (int k = 0; k < NR; ++k) {
    float tot = acc[k] + __shfl_xor(acc[k], 16, 32);
    if (i >= NS) tot = 0.0f;               // dead wave contributes nothing
    if (lane == 0) red[wave][k] = tot;
  }
  __syncthreads();
  if (tid < NR) {
    float sum = 0.0f;
    for (int w = 0; w < IPB; ++w) sum += red[w][tid];
    ws[4 + blockIdx.x * 4 + tid] = sum;
  }
}

// ---------------------------------------------------------------------------
// Kernel 3: deterministic final combine (f64 for the cancellation-heavy part)
// ---------------------------------------------------------------------------
__global__ void combine_kernel(const float* __restrict__ ws,
                               float* __restrict__ out) {
  __shared__ double contrib[NR];
  const int tid = threadIdx.x;
  if (tid < NR) {
    const double lalm = (double)ws[tid];
    double lall = 0.0;
    for (int b = 0; b < NBLOCKS; ++b) lall += (double)ws[4 + b * 4 + tid];
    const double nbB   = (double)NB_B;
    const double nbNot = (double)NS * (double)NT - nbB;
    const double lnot  = lall - lalm;
    const double lk    = lalm * nbNot + (EPSILON * nbNot - lnot) * nbB;
    const double p     = (tid == 0) ? OMEGA : RVAL;
    contrib[tid] = p * lk;
  }
  __syncthreads();
  if (tid == 0) {
    const double ret = (contrib[0] + contrib[1] + contrib[2] + contrib[3]) /
                       (double)DD / ((double)NS * (double)NT);
    out[0] = (float)ret;
  }
}

// ---------------------------------------------------------------------------
extern "C" void kernel_launch(void* const* d_in, const int* in_sizes, int n_in,
                              void* d_out, int out_size, void* d_ws,
                              size_t ws_size, hipStream_t stream) {
  const float* s    = (const float*)d_in[0];
  const float* t    = (const float*)d_in[1];
  const int*   rows = (const int*)d_in[2];
  const int*   cols = (const int*)d_in[3];
  float* ws  = (float*)d_ws;    // [0..3] loss_ALM, [4..4+NBLOCKS*4) partials
  float* out = (float*)d_out;

  anchor_kernel<<<1, 256, 0, stream>>>(s, t, rows, cols, ws);
  allpairs_kernel<<<NBLOCKS, 256, 0, stream>>>(s, t, ws);
  combine_kernel<<<1, 64, 0, stream>>>(ws, out);
}